// GIConv2dWeights_35940286332992
// MI455X (gfx1250) — compile-verified
//
#include <hip/hip_runtime.h>
#include <math.h>

// Problem constants (from reference)
#define S_    8
#define N_    32
#define CIN_  128
#define Hh_   16
#define Ww_   16
#define L_    256          // H*W
#define COUT_ 128
#define F_    1152         // CIN*3*3
#define LAM_  1152.0f
#define KTOT_ (N_ * L_)    // 8192 reduction length for both GEMMs
#define NBLK_ (F_ / 32)    // 36 Cholesky blocks

typedef __attribute__((ext_vector_type(2))) float v2f;
typedef __attribute__((ext_vector_type(8))) float v8f;

// CDNA5 async global->LDS path (ASYNCcnt-tracked), with guaranteed fallback.
#if defined(__has_builtin)
#  if __has_builtin(__builtin_amdgcn_global_load_async_to_lds_b64)
#    define HAVE_ASYNC_LDS 1
#  endif
#  if __has_builtin(__builtin_amdgcn_s_wait_asynccnt)
#    define HAVE_WAIT_ASYNC 1
#  endif
#endif

__device__ __forceinline__ void wait_asynccnt0() {
#if defined(HAVE_WAIT_ASYNC)
  __builtin_amdgcn_s_wait_asynccnt(0);
#elif defined(HAVE_ASYNC_LDS)
  asm volatile("s_wait_asynccnt 0x0" ::: "memory");
#endif
}

#if defined(HAVE_ASYNC_LDS)
// Builtin signature (from hipcc diagnostic): param0 is AS(1) pointer to
// 'int __attribute__((vector_size(8)))' (printed as '__device__' in HIP mode);
// param1 is the LDS-side pointer; args 2,3 are imm offset / cpol.
typedef int async_v2i __attribute__((vector_size(8)));
__device__ __forceinline__ void async_copy_b64(const float* g, float* l) {
  __builtin_amdgcn_global_load_async_to_lds_b64(
      (__attribute__((address_space(1))) async_v2i*)(async_v2i*)g,
      (__attribute__((address_space(3))) async_v2i*)(async_v2i*)l,
      0, 0);
}
#endif

// ---------------------------------------------------------------------------
// Prep: sqrt_prec[n] = exp(0.5*log_prec_scaled[n]); zero scalar accumulators.
// ---------------------------------------------------------------------------
__global__ void prep_kernel(const float* __restrict__ lp,
                            float* __restrict__ sqrtp,
                            float* __restrict__ scal) {
  int t = threadIdx.x;
  if (t < 32) scal[t] = 0.0f;               // [0..7]=logdet [8..15]=sumW2 [16..23]=sumZ2
  if (t < N_) sqrtp[t] = expf(0.5f * lp[t]);
}

// ---------------------------------------------------------------------------
// Yil[n,c,l] = sqrt_prec[n] * u[n,c,l]
// ---------------------------------------------------------------------------
__global__ __launch_bounds__(256) void scaley_kernel(const float* __restrict__ u,
                                                     const float* __restrict__ sqrtp,
                                                     float* __restrict__ Yil) {
  int idx = blockIdx.x * 256 + threadIdx.x;   // N*COUT*L = 1,048,576 exact
  int n = idx >> 15;                          // COUT*L = 32768
  Yil[idx] = u[idx] * sqrtp[n];
}

// ---------------------------------------------------------------------------
// im2col with per-n scaling: Xu[s,n,f,l] (f = cin*9 + ki*3 + kj, l = i*16+j)
// ---------------------------------------------------------------------------
__global__ __launch_bounds__(256) void im2col_kernel(const float* __restrict__ X,
                                                     const float* __restrict__ sqrtp,
                                                     float* __restrict__ Xu) {
  size_t idx = (size_t)blockIdx.x * 256 + threadIdx.x;   // S*N*F*L exact multiple of 256
  int l = (int)(idx & 255);
  size_t t = idx >> 8;
  int f  = (int)(t % F_);
  int sn = (int)(t / F_);          // s*N + n
  int n  = sn & (N_ - 1);
  int cin = f / 9;
  int rem = f - cin * 9;
  int ki = rem / 3;
  int kj = rem - ki * 3;
  int i = l >> 4, j = l & 15;
  int ii = i + ki - 1, jj = j + kj - 1;
  float v = 0.0f;
  if ((unsigned)ii < 16u && (unsigned)jj < 16u)
    v = X[((size_t)sn * CIN_ + cin) * 256 + ii * 16 + jj] * sqrtp[n];
  Xu[idx] = v;
}

// ---------------------------------------------------------------------------
// Generic FP32 WMMA GEMM over k=(n,l):
//   C[r,c] = sum_{n,l} A[(n*aNS + r)*256 + l] * B[(n*bNS + c)*256 + l]
// 128x128 C tile per block, 8 waves (4x2), each wave = 32x64 via 8 wmma accums.
// Double-buffered LDS; tiles staged with GLOBAL_LOAD_ASYNC_TO_LDS_B64 when
// available (ASYNCcnt), else sync float4 loads. One barrier per k-chunk.
// Exact tiling (no masks): M,Ncols multiples of 128, K=8192 multiple of 16.
// ---------------------------------------------------------------------------
__global__ __launch_bounds__(256) void wmma_gemm_nl(const float* __restrict__ Abase,
                                                    const float* __restrict__ Bbase,
                                                    float* __restrict__ Cbase,
                                                    long aSS, long bSS, long cSS,
                                                    int aNS, int bNS, int ldc, int addDiag) {
  const int tid = threadIdx.x;
  const int s = blockIdx.z;
  const float* A = Abase + (size_t)s * aSS;
  const float* B = Bbase + (size_t)s * bSS;
  float* C = Cbase + (size_t)s * cSS;
  const int row0 = blockIdx.y * 128;
  const int col0 = blockIdx.x * 128;

  // Row stride 18 floats: 8B-aligned half-rows (async B64) AND conflict-free
  // 16-lane fragment reads (18 = 2*9, 9 odd -> lh*18 mod 32 distinct).
  __shared__ float As[2][128][18];
  __shared__ float Bs[2][128][18];

  const int wid = tid >> 5, lane = tid & 31;
  const int wr = wid >> 1, wc = wid & 1;      // 4x2 wave grid
  const int lh = lane & 15;
  const bool hi = lane >= 16;

  v8f acc[2][4];
#pragma unroll
  for (int i = 0; i < 2; ++i)
#pragma unroll
    for (int j = 0; j < 4; ++j)
#pragma unroll
      for (int r = 0; r < 8; ++r) acc[i][j][r] = 0.0f;

  const int lr = tid >> 1;           // tile row loaded by this thread (0..127)
  const int lhalf = (tid & 1) * 8;   // 0 or 8 within the 16-wide k chunk

  // ---- Prologue: stage chunk 0 into buffer 0 ----
  {
    const float* ap = A + ((size_t)0 * aNS + (row0 + lr)) * 256 + lhalf;
    const float* bp = B + ((size_t)0 * bNS + (col0 + lr)) * 256 + lhalf;
#if defined(HAVE_ASYNC_LDS)
#pragma unroll
    for (int q = 0; q < 8; q += 2) {
      async_copy_b64(ap + q, &As[0][lr][lhalf + q]);
      async_copy_b64(bp + q, &Bs[0][lr][lhalf + q]);
    }
    wait_asynccnt0();
#else
    float4 a0 = *(const float4*)ap;
    float4 a1 = *(const float4*)(ap + 4);
    float4 b0 = *(const float4*)bp;
    float4 b1 = *(const float4*)(bp + 4);
    As[0][lr][lhalf + 0] = a0.x; As[0][lr][lhalf + 1] = a0.y;
    As[0][lr][lhalf + 2] = a0.z; As[0][lr][lhalf + 3] = a0.w;
    As[0][lr][lhalf + 4] = a1.x; As[0][lr][lhalf + 5] = a1.y;
    As[0][lr][lhalf + 6] = a1.z; As[0][lr][lhalf + 7] = a1.w;
    Bs[0][lr][lhalf + 0] = b0.x; Bs[0][lr][lhalf + 1] = b0.y;
    Bs[0][lr][lhalf + 2] = b0.z; Bs[0][lr][lhalf + 3] = b0.w;
    Bs[0][lr][lhalf + 4] = b1.x; Bs[0][lr][lhalf + 5] = b1.y;
    Bs[0][lr][lhalf + 6] = b1.z; Bs[0][lr][lhalf + 7] = b1.w;
#endif
  }
  __syncthreads();

  int bsel = 0;
  for (int k0 = 0; k0 < KTOT_; k0 += 16, bsel ^= 1) {
    const int nxt = bsel ^ 1;
    const bool more = (k0 + 16) < KTOT_;     // uniform across block

    // ---- Prefetch next chunk into the other buffer ----
#if !defined(HAVE_ASYNC_LDS)
    float4 pa0, pa1, pb0, pb1;
#endif
    if (more) {
      const int n2 = (k0 + 16) >> 8;
      const int l2 = (k0 + 16) & 255;
      const float* ap = A + ((size_t)n2 * aNS + (row0 + lr)) * 256 + l2 + lhalf;
      const float* bp = B + ((size_t)n2 * bNS + (col0 + lr)) * 256 + l2 + lhalf;
#if defined(HAVE_ASYNC_LDS)
#pragma unroll
      for (int q = 0; q < 8; q += 2) {
        async_copy_b64(ap + q, &As[nxt][lr][lhalf + q]);
        async_copy_b64(bp + q, &Bs[nxt][lr][lhalf + q]);
      }
#else
      pa0 = *(const float4*)ap;
      pa1 = *(const float4*)(ap + 4);
      pb0 = *(const float4*)bp;
      pb1 = *(const float4*)(bp + 4);
#endif
    }

    // ---- Compute on current buffer ----
#pragma unroll
    for (int kk = 0; kk < 16; kk += 4) {
      const int kb = kk + (hi ? 2 : 0);
      // A fragments per ISA: lanes0-15 hold {K0,K1} of row M=lane; lanes16-31 {K2,K3}
      v2f a0, a1;
      a0.x = As[bsel][wr * 32 + lh][kb];      a0.y = As[bsel][wr * 32 + lh][kb + 1];
      a1.x = As[bsel][wr * 32 + 16 + lh][kb]; a1.y = As[bsel][wr * 32 + 16 + lh][kb + 1];
      // B fragments mirror A (K split across lane halves, N=lane within half)
      v2f b0, b1, b2, b3;
      b0.x = Bs[bsel][wc * 64 +  0 + lh][kb]; b0.y = Bs[bsel][wc * 64 +  0 + lh][kb + 1];
      b1.x = Bs[bsel][wc * 64 + 16 + lh][kb]; b1.y = Bs[bsel][wc * 64 + 16 + lh][kb + 1];
      b2.x = Bs[bsel][wc * 64 + 32 + lh][kb]; b2.y = Bs[bsel][wc * 64 + 32 + lh][kb + 1];
      b3.x = Bs[bsel][wc * 64 + 48 + lh][kb]; b3.y = Bs[bsel][wc * 64 + 48 + lh][kb + 1];
      acc[0][0] = __builtin_amdgcn_wmma_f32_16x16x4_f32(false, a0, false, b0, (short)0, acc[0][0], false, false);
      acc[0][1] = __builtin_amdgcn_wmma_f32_16x16x4_f32(false, a0, false, b1, (short)0, acc[0][1], false, false);
      acc[0][2] = __builtin_amdgcn_wmma_f32_16x16x4_f32(false, a0, false, b2, (short)0, acc[0][2], false, false);
      acc[0][3] = __builtin_amdgcn_wmma_f32_16x16x4_f32(false, a0, false, b3, (short)0, acc[0][3], false, false);
      acc[1][0] = __builtin_amdgcn_wmma_f32_16x16x4_f32(false, a1, false, b0, (short)0, acc[1][0], false, false);
      acc[1][1] = __builtin_amdgcn_wmma_f32_16x16x4_f32(false, a1, false, b1, (short)0, acc[1][1], false, false);
      acc[1][2] = __builtin_amdgcn_wmma_f32_16x16x4_f32(false, a1, false, b2, (short)0, acc[1][2], false, false);
      acc[1][3] = __builtin_amdgcn_wmma_f32_16x16x4_f32(false, a1, false, b3, (short)0, acc[1][3], false, false);
    }

    // ---- Publish prefetched chunk, single barrier per iteration ----
#if defined(HAVE_ASYNC_LDS)
    wait_asynccnt0();
#else
    if (more) {
      As[nxt][lr][lhalf + 0] = pa0.x; As[nxt][lr][lhalf + 1] = pa0.y;
      As[nxt][lr][lhalf + 2] = pa0.z; As[nxt][lr][lhalf + 3] = pa0.w;
      As[nxt][lr][lhalf + 4] = pa1.x; As[nxt][lr][lhalf + 5] = pa1.y;
      As[nxt][lr][lhalf + 6] = pa1.z; As[nxt][lr][lhalf + 7] = pa1.w;
      Bs[nxt][lr][lhalf + 0] = pb0.x; Bs[nxt][lr][lhalf + 1] = pb0.y;
      Bs[nxt][lr][lhalf + 2] = pb0.z; Bs[nxt][lr][lhalf + 3] = pb0.w;
      Bs[nxt][lr][lhalf + 4] = pb1.x; Bs[nxt][lr][lhalf + 5] = pb1.y;
      Bs[nxt][lr][lhalf + 6] = pb1.z; Bs[nxt][lr][lhalf + 7] = pb1.w;
    }
#endif
    __syncthreads();
  }

  // Epilogue: C/D layout — VGPR r: lanes0-15 -> (M=r, N=lane), lanes16-31 -> (M=8+r, N=lane-16)
#pragma unroll
  for (int i = 0; i < 2; ++i) {
#pragma unroll
    for (int j = 0; j < 4; ++j) {
      int rbase = row0 + wr * 32 + i * 16 + (hi ? 8 : 0);
      int cidx  = col0 + wc * 64 + j * 16 + lh;
#pragma unroll
      for (int r = 0; r < 8; ++r) {
        float v = acc[i][j][r];
        int row = rbase + r;
        if (addDiag && row == cidx) v += LAM_;   // prec = XLX + LAM*I
        C[(size_t)row * ldc + cidx] = v;
      }
    }
  }
}

// ---------------------------------------------------------------------------
// Blocked in-place Cholesky (lower), NB=32, one workgroup per s.
// Also produces logdet[s] = 2*sum(log diag L).
// ---------------------------------------------------------------------------
__global__ __launch_bounds__(256) void cholesky_kernel(float* __restrict__ Pm,
                                                       float* __restrict__ logdet) {
  const int s = blockIdx.x, tid = threadIdx.x;
  float* M = Pm + (size_t)s * F_ * F_;
  __shared__ float Ds[32][33];
  __shared__ float Lk[32][33];
  float ldsum = 0.0f;

  for (int jb = 0; jb < NBLK_; ++jb) {
    const int c0 = jb * 32;
    __syncthreads();
    for (int e = tid; e < 1024; e += 256) {
      int i = e >> 5, j = e & 31;
      Ds[i][j] = M[(size_t)(c0 + i) * F_ + c0 + j];
    }
    __syncthreads();
    // Micro-factor 32x32 diagonal block in LDS
    for (int kk = 0; kk < 32; ++kk) {
      if (tid == 0) Ds[kk][kk] = sqrtf(Ds[kk][kk]);
      __syncthreads();
      if (tid > kk && tid < 32) Ds[tid][kk] /= Ds[kk][kk];
      __syncthreads();
      if (tid > kk && tid < 32) {
        float lik = Ds[tid][kk];
        for (int m = kk + 1; m <= tid; ++m) Ds[tid][m] -= lik * Ds[m][kk];
      }
      __syncthreads();
    }
    if (tid == 0) {
      for (int i = 0; i < 32; ++i) ldsum += logf(Ds[i][i]);
    }
    if (tid < 32) {
      for (int j = 0; j <= tid; ++j) M[(size_t)(c0 + tid) * F_ + c0 + j] = Ds[tid][j];
    }
    __syncthreads();
    // Panel solve: rows below diag block, x = row * inv(Ds^T)
    for (int i = c0 + 32 + tid; i < F_; i += 256) {
      float x[32];
      const float* row = M + (size_t)i * F_ + c0;
#pragma unroll
      for (int kk = 0; kk < 32; ++kk) x[kk] = row[kk];
#pragma unroll
      for (int kk = 0; kk < 32; ++kk) {
        float v = x[kk];
#pragma unroll
        for (int m = 0; m < kk; ++m) v -= x[m] * Ds[kk][m];
        x[kk] = v / Ds[kk][kk];
      }
      float* wrow = M + (size_t)i * F_ + c0;
#pragma unroll
      for (int kk = 0; kk < 32; ++kk) wrow[kk] = x[kk];
    }
    __syncthreads();
    // Trailing update (lower part only): P[i, kb-block] -= Li . Lk^T
    for (int kb = jb + 1; kb < NBLK_; ++kb) {
      __syncthreads();
      for (int e = tid; e < 1024; e += 256) {
        int r = e >> 5, m = e & 31;
        Lk[r][m] = M[(size_t)(kb * 32 + r) * F_ + c0 + m];
      }
      __syncthreads();
      for (int i = kb * 32 + tid; i < F_; i += 256) {
        float Li[32];
        const float* prow = M + (size_t)i * F_ + c0;
#pragma unroll
        for (int m = 0; m < 32; ++m) Li[m] = prow[m];
        int rel = i - kb * 32;
        int mmax = rel < 31 ? rel + 1 : 32;
        float* urow = M + (size_t)i * F_ + kb * 32;
        for (int m = 0; m < mmax; ++m) {
          float d = 0.0f;
#pragma unroll
          for (int t = 0; t < 32; ++t) d += Li[t] * Lk[m][t];
          urow[m] -= d;
        }
      }
    }
    __syncthreads();
  }
  if (tid == 0) logdet[s] = 2.0f * ldsum;
}

// ---------------------------------------------------------------------------
// Forward solve: T1 = L^{-1} * XLY  (one thread per RHS column, blocked NB=32)
// ---------------------------------------------------------------------------
__global__ __launch_bounds__(128) void fwd_solve_kernel(const float* __restrict__ P,
                                                        const float* __restrict__ Bmat,
                                                        float* __restrict__ T1) {
  const int s = blockIdx.x, c = threadIdx.x;
  const float* M = P + (size_t)s * F_ * F_;
  const float* Bm = Bmat + (size_t)s * F_ * COUT_;
  float* T = T1 + (size_t)s * F_ * COUT_;
  __shared__ float Lb[32][33];

  for (int b = 0; b < NBLK_; ++b) {
    const int r0 = b * 32;
    float acc[32];
#pragma unroll
    for (int i = 0; i < 32; ++i) acc[i] = Bm[(size_t)(r0 + i) * COUT_ + c];
    for (int jb = 0; jb < b; ++jb) {
      __syncthreads();
      for (int e = c; e < 1024; e += 128) {
        int i = e >> 5, j = e & 31;
        Lb[i][j] = M[(size_t)(r0 + i) * F_ + jb * 32 + j];
      }
      __syncthreads();
      for (int jj = 0; jj < 32; ++jj) {
        float t = T[(size_t)(jb * 32 + jj) * COUT_ + c];
#pragma unroll
        for (int i = 0; i < 32; ++i) acc[i] -= Lb[i][jj] * t;
      }
    }
    __syncthreads();
    for (int e = c; e < 1024; e += 128) {
      int i = e >> 5, j = e & 31;
      Lb[i][j] = M[(size_t)(r0 + i) * F_ + r0 + j];
    }
    __syncthreads();
#pragma unroll
    for (int i = 0; i < 32; ++i) {
      float v = acc[i];
#pragma unroll
      for (int m = 0; m < i; ++m) v -= Lb[i][m] * acc[m];
      acc[i] = v / Lb[i][i];
    }
#pragma unroll
    for (int i = 0; i < 32; ++i) T[(size_t)(r0 + i) * COUT_ + c] = acc[i];
    __syncthreads();
  }
}

// ---------------------------------------------------------------------------
// Backward solve: W = L^{-T} (T1 + Z).  Writes sample (transposed) + sum(W^2).
// ---------------------------------------------------------------------------
__global__ __launch_bounds__(128) void bwd_solve_kernel(const float* __restrict__ P,
                                                        const float* __restrict__ T1,
                                                        const float* __restrict__ Z,
                                                        float* __restrict__ Wbuf,
                                                        float* __restrict__ outSample,
                                                        float* __restrict__ sumW2) {
  const int s = blockIdx.x, c = threadIdx.x;
  const float* M = P + (size_t)s * F_ * F_;
  const float* T = T1 + (size_t)s * F_ * COUT_;
  const float* Zs = Z + (size_t)s * F_ * COUT_;
  float* Wm = Wbuf + (size_t)s * F_ * COUT_;
  float* Os = outSample + (size_t)s * COUT_ * F_;
  __shared__ float Lb[32][33];
  float ssq = 0.0f;

  for (int b = NBLK_ - 1; b >= 0; --b) {
    const int r0 = b * 32;
    float acc[32];
#pragma unroll
    for (int i = 0; i < 32; ++i)
      acc[i] = T[(size_t)(r0 + i) * COUT_ + c] + Zs[(size_t)(r0 + i) * COUT_ + c];
    for (int jb = b + 1; jb < NBLK_; ++jb) {
      __syncthreads();
      for (int e = c; e < 1024; e += 128) {
        int jj = e >> 5, i = e & 31;
        Lb[jj][i] = M[(size_t)(jb * 32 + jj) * F_ + r0 + i];   // L[j, r0+i] = U[r0+i, j]
      }
      __syncthreads();
      for (int jj = 0; jj < 32; ++jj) {
        float wv = Wm[(size_t)(jb * 32 + jj) * COUT_ + c];
#pragma unroll
        for (int i = 0; i < 32; ++i) acc[i] -= Lb[jj][i] * wv;
      }
    }
    __syncthreads();
    for (int e = c; e < 1024; e += 128) {
      int i = e >> 5, j = e & 31;
      Lb[i][j] = M[(size_t)(r0 + i) * F_ + r0 + j];
    }
    __syncthreads();
#pragma unroll
    for (int i = 31; i >= 0; --i) {
      float v = acc[i];
#pragma unroll
      for (int m = i + 1; m < 32; ++m) v -= Lb[m][i] * acc[m];   // U[i,m] = L[m,i]
      acc[i] = v / Lb[i][i];
    }
#pragma unroll
    for (int i = 0; i < 32; ++i) {
      Wm[(size_t)(r0 + i) * COUT_ + c] = acc[i];
      Os[(size_t)c * F_ + r0 + i] = acc[i];                      // sample[s, c, f]
      ssq += acc[i] * acc[i];
    }
    __syncthreads();
  }
  atomicAdd(&sumW2[s], ssq);
}

// ---------------------------------------------------------------------------
// sum(Z^2) per s
// ---------------------------------------------------------------------------
__global__ __launch_bounds__(256) void z2_kernel(const float* __restrict__ Z,
                                                 float* __restrict__ sumZ2) {
  const int s = blockIdx.y;
  const float* Zs = Z + (size_t)s * F_ * COUT_;
  const int n = F_ * COUT_;
  float p = 0.0f;
  for (int i = blockIdx.x * 256 + threadIdx.x; i < n; i += gridDim.x * 256) {
    float z = Zs[i];
    p += z * z;
  }
  __shared__ float red[256];
  red[threadIdx.x] = p;
  __syncthreads();
  for (int off = 128; off > 0; off >>= 1) {
    if (threadIdx.x < off) red[threadIdx.x] += red[threadIdx.x + off];
    __syncthreads();
  }
  if (threadIdx.x == 0) atomicAdd(&sumZ2[s], red[0]);
}

// ---------------------------------------------------------------------------
// logpq[s] = logP - logQ
// ---------------------------------------------------------------------------
__global__ void final_kernel(const float* __restrict__ scal, float* __restrict__ outLogpq) {
  int s = threadIdx.x;
  if (s < S_) {
    float logdet = scal[s];
    float sW2 = scal[8 + s];
    float sZ2 = scal[16 + s];
    float logP = -0.5f * LAM_ * sW2 + 0.5f * (float)(COUT_ * F_) * logf(LAM_);
    float logQ = -0.5f * sZ2 + 0.5f * (float)COUT_ * logdet;
    outLogpq[s] = logP - logQ;
  }
}

// ---------------------------------------------------------------------------
extern "C" void kernel_launch(void* const* d_in, const int* in_sizes, int n_in,
                              void* d_out, int out_size, void* d_ws, size_t ws_size,
                              hipStream_t stream) {
  (void)in_sizes; (void)n_in; (void)out_size; (void)ws_size;
  const float* X  = (const float*)d_in[0];   // [S,N,CIN,H,W]
  const float* u  = (const float*)d_in[1];   // [N,COUT,H,W]
  const float* lp = (const float*)d_in[2];   // [N]
  const float* Z  = (const float*)d_in[3];   // [S,F,COUT]
  float* out = (float*)d_out;
  float* ws  = (float*)d_ws;

  // Workspace layout (floats), all offsets 16-float aligned
  float* sqrtp = ws;                                     // 32
  float* scal  = ws + 64;                                // 32 scalars
  float* Yil   = ws + 128;                               // N*COUT*L   = 1,048,576
  float* XLYw  = Yil  + (size_t)N_ * COUT_ * L_;         // S*F*COUT   = 1,179,648
  float* T1    = XLYw + (size_t)S_ * F_ * COUT_;         // S*F*COUT
  float* Wbuf  = T1   + (size_t)S_ * F_ * COUT_;         // S*F*COUT
  float* P     = Wbuf + (size_t)S_ * F_ * COUT_;         // S*F*F      = 10,616,832
  float* Xu    = P    + (size_t)S_ * F_ * F_;            // S*N*F*L    = 75,497,472
  // total ~ 363 MB of workspace

  prep_kernel<<<1, 64, 0, stream>>>(lp, sqrtp, scal);
  scaley_kernel<<<(N_ * COUT_ * L_) / 256, 256, 0, stream>>>(u, sqrtp, Yil);
  im2col_kernel<<<(int)(((size_t)S_ * N_ * F_ * L_) / 256), 256, 0, stream>>>(X, sqrtp, Xu);

  // XLX -> P (with +LAM*I): C[1152x1152] per s, A=B=Xu
  wmma_gemm_nl<<<dim3(F_ / 128, F_ / 128, S_), 256, 0, stream>>>(
      Xu, Xu, P,
      (long)N_ * F_ * L_, (long)N_ * F_ * L_, (long)F_ * F_,
      F_, F_, F_, 1);

  // XLY: C[1152x128] per s, B = Yil (no s stride)
  wmma_gemm_nl<<<dim3(COUT_ / 128, F_ / 128, S_), 256, 0, stream>>>(
      Xu, Yil, XLYw,
      (long)N_ * F_ * L_, 0L, (long)F_ * COUT_,
      F_, COUT_, COUT_, 0);

  cholesky_kernel<<<S_, 256, 0, stream>>>(P, scal);          // scal[0..7] = logdet
  fwd_solve_kernel<<<S_, 128, 0, stream>>>(P, XLYw, T1);
  z2_kernel<<<dim3(72, S_), 256, 0, stream>>>(Z, scal + 16); // scal[16..23] = sumZ2
  bwd_solve_kernel<<<S_, 128, 0, stream>>>(P, T1, Z, Wbuf, out, scal + 8);
  final_kernel<<<1, 32, 0, stream>>>(scal, out + (size_t)S_ * COUT_ * F_);
}